// NeuralSlamMemoryCore_68728066671078
// MI455X (gfx1250) — compile-verified
//
#include <hip/hip_runtime.h>
#include <hip/hip_bf16.h>
#include <math.h>

#define BN   2048
#define CCH  4
#define HWN  4096   // 64*64
#define PW   66     // padded width (64 + 1 halo each side)
#define PSZ  (PW*PW)

static constexpr size_t OFF_W = (size_t)BN * CCH * HWN;          // weights_2d offset
static constexpr size_t OFF_R = OFF_W + (size_t)BN * HWN;        // read_vector offset

typedef float v2f __attribute__((ext_vector_type(2)));
typedef float v8f __attribute__((ext_vector_type(8)));

__device__ __forceinline__ float softplusf(float x) {
    return fmaxf(x, 0.f) + log1pf(__expf(-fabsf(x)));
}
__device__ __forceinline__ float sigmoidf(float x) {
    return 1.f / (1.f + __expf(-x));
}
__device__ __forceinline__ float wave_sum(float v) {
#pragma unroll
    for (int off = 16; off > 0; off >>= 1) v += __shfl_xor(v, off, 32);
    return v;
}
__device__ __forceinline__ float wave_max(float v) {
#pragma unroll
    for (int off = 16; off > 0; off >>= 1) v = fmaxf(v, __shfl_xor(v, off, 32));
    return v;
}
__device__ __forceinline__ float block_sum(float v, float* s_red, float* s_bc, int tid) {
    v = wave_sum(v);
    if ((tid & 31) == 0) s_red[tid >> 5] = v;
    __syncthreads();
    if (tid == 0) {
        float t = 0.f;
#pragma unroll
        for (int i = 0; i < 8; ++i) t += s_red[i];
        s_bc[0] = t;
    }
    __syncthreads();
    float r = s_bc[0];
    __syncthreads();
    return r;
}
__device__ __forceinline__ float block_max(float v, float* s_red, float* s_bc, int tid) {
    v = wave_max(v);
    if ((tid & 31) == 0) s_red[tid >> 5] = v;
    __syncthreads();
    if (tid == 0) {
        float t = -INFINITY;
#pragma unroll
        for (int i = 0; i < 8; ++i) t = fmaxf(t, s_red[i]);
        s_bc[0] = t;
    }
    __syncthreads();
    float r = s_bc[0];
    __syncthreads();
    return r;
}

__global__ __launch_bounds__(256)
void ntm_core_kernel(const float* __restrict__ map_state,
                     const float* __restrict__ weight_state,
                     const float* __restrict__ key,
                     const float* __restrict__ beta,
                     const float* __restrict__ gate,
                     const float* __restrict__ shift,
                     const float* __restrict__ sharpen,
                     const float* __restrict__ add_patch_logits,
                     const float* __restrict__ erase_patch_logits,
                     float* __restrict__ out)
{
    __shared__ float w_a[HWN];      // gated weights (flat 64x64)
    __shared__ float w_pad[PSZ];    // final weights, zero-padded 66x66
    __shared__ float s_P[16][12];   // WMMA A-matrix: rows 0-3 add patch, 8-11 erase patch
    __shared__ float s_shift[9];
    __shared__ float s_kn[4];
    __shared__ float s_sc[4];       // beta_pos, gate_sigma, sharpen_pos
    __shared__ float s_red[8];
    __shared__ float s_bc[1];
    __shared__ float s_rv[4];

    const int b   = blockIdx.x;
    const int tid = threadIdx.x;

    // ---- Phase 0: per-batch scalars + zero the padded weight buffer ----
    for (int idx = tid; idx < PSZ; idx += 256) w_pad[idx] = 0.f;

    if (tid == 0) {
        float k0 = key[b*4+0], k1 = key[b*4+1], k2 = key[b*4+2], k3 = key[b*4+3];
        float nrm = sqrtf(k0*k0 + k1*k1 + k2*k2 + k3*k3);
        float inv = 1.f / fmaxf(nrm, 1e-12f);
        s_kn[0] = k0*inv; s_kn[1] = k1*inv; s_kn[2] = k2*inv; s_kn[3] = k3*inv;
        s_sc[0] = softplusf(beta[b]) + 1e-6f;
        s_sc[1] = sigmoidf(gate[b]);
        s_sc[2] = 1.f + softplusf(sharpen[b]);
        float m = -INFINITY;
        for (int j = 0; j < 9; ++j) m = fmaxf(m, shift[b*9+j]);
        float s = 0.f, e[9];
        for (int j = 0; j < 9; ++j) { e[j] = __expf(shift[b*9+j] - m); s += e[j]; }
        float is = 1.f / s;
        for (int j = 0; j < 9; ++j) s_shift[j] = e[j] * is;
    }
    if (tid < 4) s_rv[tid] = 0.f;
    if (tid < 192) {                       // 16x12 padded patch matrix, one writer per entry
        int m = tid / 12, j = tid % 12;
        float v = 0.f;
        if (j < 9) {
            if (m < 4)                  v = sigmoidf(add_patch_logits  [(size_t)b*36 + m*9 + j]);
            else if (m >= 8 && m < 12)  v = sigmoidf(erase_patch_logits[(size_t)b*36 + (m-8)*9 + j]);
        }
        s_P[m][j] = v;
    }
    __syncthreads();

    const float kn0 = s_kn[0], kn1 = s_kn[1], kn2 = s_kn[2], kn3 = s_kn[3];
    const float betap = s_sc[0], gsig = s_sc[1], sharp = s_sc[2];
    const size_t mbase = (size_t)b * (CCH * HWN);

    // ---- Phase 1: cosine similarity -> softmax -> gate with prev weights ----
    float tv[16];
    float lmax = -INFINITY;
#pragma unroll
    for (int i = 0; i < 16; ++i) {
        int p = tid + i * 256;
        float m0 = map_state[mbase + p];
        float m1 = map_state[mbase + HWN + p];
        float m2 = map_state[mbase + 2*HWN + p];
        float m3 = map_state[mbase + 3*HWN + p];
        float nrm = sqrtf(m0*m0 + m1*m1 + m2*m2 + m3*m3);
        float inv = 1.f / fmaxf(nrm, 1e-12f);
        float sim = (m0*kn0 + m1*kn1 + m2*kn2 + m3*kn3) * inv;
        tv[i] = betap * sim;
        lmax = fmaxf(lmax, tv[i]);
    }
    float tmax = block_max(lmax, s_red, s_bc, tid);
    float lsum = 0.f;
#pragma unroll
    for (int i = 0; i < 16; ++i) { tv[i] = __expf(tv[i] - tmax); lsum += tv[i]; }
    float ssum = block_sum(lsum, s_red, s_bc, tid);
    float isum = 1.f / ssum;
#pragma unroll
    for (int i = 0; i < 16; ++i) {
        int p = tid + i * 256;
        float content = tv[i] * isum;
        float prev = weight_state[(size_t)b * HWN + p];
        w_a[p] = gsig * content + (1.f - gsig) * prev;
    }
    __syncthreads();

    // ---- Phase 2: circular 3x3 shift conv + sharpen + analytic double-normalize ----
    float lsum2 = 0.f;
#pragma unroll
    for (int i = 0; i < 16; ++i) {
        int p = tid + i * 256;
        int h = p >> 6, wc = p & 63;
        float acc = 0.f;
#pragma unroll
        for (int dh = 0; dh < 3; ++dh) {
            int hh = (h + 1 - dh + 64) & 63;
#pragma unroll
            for (int dw = 0; dw < 3; ++dw) {
                int ww = (wc + 1 - dw + 64) & 63;
                acc += s_shift[dh*3 + dw] * w_a[hh*64 + ww];
            }
        }
        float x = fmaxf(acc, 1e-6f);
        tv[i] = powf(x, sharp);
        lsum2 += tv[i];
    }
    float S1 = block_sum(lsum2, s_red, s_bc, tid);
    float S2 = S1 / (S1 + 1e-6f);
    float scale = 1.f / ((S1 + 1e-6f) * (S2 + 1e-6f));
#pragma unroll
    for (int i = 0; i < 16; ++i) {
        int p = tid + i * 256;
        int h = p >> 6, wc = p & 63;
        float wf = tv[i] * scale;
        w_pad[(h + 1) * PW + (wc + 1)] = wf;     // interior of padded buffer
        out[OFF_W + (size_t)b * HWN + p] = wf;
    }
    __syncthreads();

    // ---- Phase 3: WMMA patch apply (add+erase), map update, read vector ----
    // D = P(16x12) * S(12x16) via 3 x V_WMMA_F32_16X16X4_F32 accumulating.
    // A layout (32-bit 16x4): lanes 0-15 -> (M=lane, K=0/1 in vgpr0/1); lanes 16-31 -> K=2/3.
    const int lane = tid & 31;
    const int wid  = __builtin_amdgcn_readfirstlane(tid >> 5);   // provably uniform
    const int n    = lane & 15;
    const bool hi  = lane >= 16;
    const int kh   = hi ? 2 : 0;

    v2f a0, a1, a2;
    a0.x = s_P[n][0 + kh]; a0.y = s_P[n][1 + kh];
    a1.x = s_P[n][4 + kh]; a1.y = s_P[n][5 + kh];
    a2.x = s_P[n][8 + kh]; a2.y = s_P[n][9 + kh];

    // tap offsets (relative to wbase) for this lane's four B-fragment slots
    // j -> (dh=j/3, dw=j%3), offset = -(dh*PW + dw);  j0..j3 = kh+0, kh+1, kh+4, kh+5
    const int off0 = -((0 + kh) / 3) * PW - ((0 + kh) % 3);
    const int off1 = -((1 + kh) / 3) * PW - ((1 + kh) % 3);
    const int off2 = -((4 + kh) / 3) * PW - ((4 + kh) % 3);
    const int off3 = -((5 + kh) / 3) * PW - ((5 + kh) % 3);

    const int cbase = hi ? 2 : 0;       // this lane updates channels cbase, cbase+1
    float racc0 = 0.f, racc1 = 0.f;

    for (int chunk = wid; chunk < 256; chunk += 8) {
        int p = chunk * 16 + n;
        int h = p >> 6, wc = p & 63;
        int wbase = (h + 2) * PW + (wc + 2);     // padded index of tap (dh=0,dw=0)

        v2f b0, b1, b2;
        b0.x = w_pad[wbase + off0];
        b0.y = w_pad[wbase + off1];
        b1.x = w_pad[wbase + off2];
        b1.y = w_pad[wbase + off3];
        // slot j = 8+kh: real tap only for low half (j=8, dh=2,dw=2); high half (j=10) is pad
        float tap8 = w_pad[wbase - 2 * PW - 2];  // unconditional load, select below
        b2.x = hi ? 0.f : tap8;
        b2.y = 0.f;                              // j=9 / j=11 are always pad

        v8f c = {0.f, 0.f, 0.f, 0.f, 0.f, 0.f, 0.f, 0.f};
        c = __builtin_amdgcn_wmma_f32_16x16x4_f32(false, a0, false, b0, (short)0, c, false, false);
        c = __builtin_amdgcn_wmma_f32_16x16x4_f32(false, a1, false, b1, (short)0, c, false, false);
        c = __builtin_amdgcn_wmma_f32_16x16x4_f32(false, a2, false, b2, (short)0, c, false, false);

        // lanes 0-15 hold add_map rows 0-3 in c[0..3]; lanes 16-31 hold erase rows (M=8..11)
        float o0 = __shfl_xor(c[0], 16, 32);
        float o1 = __shfl_xor(c[1], 16, 32);
        float o2 = __shfl_xor(c[2], 16, 32);
        float o3 = __shfl_xor(c[3], 16, 32);

        float addv0, addv1, erv0, erv1;
        if (!hi) { addv0 = c[0]; addv1 = c[1]; erv0 = o0;  erv1 = o1; }   // channels 0,1
        else     { addv0 = o2;  addv1 = o3;  erv0 = c[2]; erv1 = c[3]; }  // channels 2,3

        erv0 = fminf(fmaxf(erv0, 0.f), 1.f);
        erv1 = fminf(fmaxf(erv1, 0.f), 1.f);

        float wfin = w_pad[wbase - PW - 1];      // == weights[h][wc]
        size_t gidx = mbase + (size_t)cbase * HWN + p;
        float m0 = map_state[gidx];
        float m1 = map_state[gidx + HWN];
        float u0 = m0 * (1.f - erv0) + addv0;
        float u1 = m1 * (1.f - erv1) + addv1;
        out[gidx]       = u0;
        out[gidx + HWN] = u1;
        racc0 += u0 * wfin;
        racc1 += u1 * wfin;
    }
    atomicAdd(&s_rv[cbase],     racc0);
    atomicAdd(&s_rv[cbase + 1], racc1);
    __syncthreads();
    if (tid < 4) out[OFF_R + (size_t)b * 4 + tid] = s_rv[tid];
}

extern "C" void kernel_launch(void* const* d_in, const int* in_sizes, int n_in,
                              void* d_out, int out_size, void* d_ws, size_t ws_size,
                              hipStream_t stream) {
    const float* map_state    = (const float*)d_in[0];
    const float* weight_state = (const float*)d_in[1];
    const float* key          = (const float*)d_in[2];
    const float* beta         = (const float*)d_in[3];
    const float* gate         = (const float*)d_in[4];
    const float* shift        = (const float*)d_in[5];
    const float* sharpen      = (const float*)d_in[6];
    // d_in[7] = erase, d_in[8] = add : unused by the reference computation
    const float* add_patch    = (const float*)d_in[9];
    const float* erase_patch  = (const float*)d_in[10];
    float* out = (float*)d_out;

    hipLaunchKernelGGL(ntm_core_kernel, dim3(BN), dim3(256), 0, stream,
                       map_state, weight_state, key, beta, gate, shift, sharpen,
                       add_patch, erase_patch, out);
}